// ConnectionsFullEncode_83760452206685
// MI455X (gfx1250) — compile-verified
//
#include <hip/hip_runtime.h>

// ============================================================================
// ConnectionsFullEncode, MI455X (gfx1250, wave32)
//
// Math:  MC[c][f]   = sum_h W2[c&3][h] * W1[h][f + (c>>2)*F]   (rows 0..7 of 16x2048, 8..15 zero-pad)
//        P[row][c]  = sum_f V[row][f] * MC[c][f]               (8192 x 8)
//        const[c]   = sum_h W2[c][h]*b1[h] + b2[c]
//        out = [ V (copy) | sigmoid(P[b,i,c] + P[b,j,4+c] + const[c]) as (B,C,j,i) ]
//
// Bandwidth-bound: ~285 MB total traffic -> ~12.5 us floor at 23.3 TB/s.
// ============================================================================

typedef float v2f  __attribute__((ext_vector_type(2)));
typedef float v4f  __attribute__((ext_vector_type(4)));
typedef float v8f  __attribute__((ext_vector_type(8)));
typedef int   vsi4 __attribute__((__vector_size__(4 * sizeof(int))));

#define BATCH 8
#define SEQ   1024
#define FDIM  2048
#define HDIM  1024
#define CDIM  4
#define ROWS  (BATCH * SEQ)              // 8192
#define VEC_ELEMS ((size_t)ROWS * FDIM)  // 16777216 floats

// workspace layout (float offsets); needs (32784 + 65536)*4 = ~394 KB
#define MC_OFF    0                      // 16 x 2048 (rows 8..15 stay zero)
#define CONST_OFF (16 * FDIM)            // 32768
#define P_OFF     (CONST_OFF + 16)       // 32784
#define WS_FLOATS (P_OFF + ROWS * 8)

#if defined(__gfx1250__)
  #if __has_builtin(__builtin_amdgcn_global_load_async_to_lds_b128)
    #define HAVE_ASYNC 1
  #else
    #define HAVE_ASYNC 0
    #warning "gfx1250: __builtin_amdgcn_global_load_async_to_lds_b128 unavailable; using ds_store staging fallback"
  #endif
#else
  #define HAVE_ASYNC 0
#endif

#define AS1 __attribute__((address_space(1)))
#define AS3 __attribute__((address_space(3)))

// ---------------------------------------------------------------------------
// Kernel 0: zero MC (16 rows) + P accumulators, compute const[c] = W2[c]·b1+b2
// ---------------------------------------------------------------------------
__global__ __launch_bounds__(256) void k_init(float* __restrict__ ws,
                                              const float* __restrict__ W2,
                                              const float* __restrict__ b1,
                                              const float* __restrict__ b2) {
    int t = blockIdx.x * 256 + threadIdx.x;
    int stride = gridDim.x * 256;
    for (int i = t; i < 16 * FDIM; i += stride) ws[MC_OFF + i] = 0.0f;
    for (int i = t; i < ROWS * 8; i += stride) ws[P_OFF + i] = 0.0f;
    if (t < CDIM) {
        float acc = b2[t];
        for (int h = 0; h < HDIM; ++h) acc += W2[t * HDIM + h] * b1[h];
        ws[CONST_OFF + t] = acc;
    }
}

// ---------------------------------------------------------------------------
// Kernel 1: MC = W2 @ [W1a | W1b]  (8 x 2048), h-split x4 with f32 atomics.
// W1 is read exactly once (16 MiB), coalesced across f.
// ---------------------------------------------------------------------------
__global__ __launch_bounds__(256) void k_mc(float* __restrict__ ws,
                                            const float* __restrict__ W1,
                                            const float* __restrict__ W2) {
    int t = blockIdx.x * 256 + threadIdx.x;      // 16384 threads
    int hs   = t >> 12;                          // h-chunk 0..3
    int r    = t & 4095;
    int half = r >> 11;                          // 0 -> W1a, 1 -> W1b
    int f    = r & 2047;
    const float* w1col = W1 + (size_t)half * FDIM + f;
    float a0 = 0.f, a1 = 0.f, a2 = 0.f, a3 = 0.f;
    int h0 = hs * 256;
    for (int h = h0; h < h0 + 256; ++h) {
        float w = w1col[(size_t)h * (2 * FDIM)];
        a0 += W2[0 * HDIM + h] * w;
        a1 += W2[1 * HDIM + h] * w;
        a2 += W2[2 * HDIM + h] * w;
        a3 += W2[3 * HDIM + h] * w;
    }
    float* mc = ws + MC_OFF + (size_t)(half * 4) * FDIM + f;
    atomicAdd(mc + 0 * FDIM, a0);
    atomicAdd(mc + 1 * FDIM, a1);
    atomicAdd(mc + 2 * FDIM, a2);
    atomicAdd(mc + 3 * FDIM, a3);
}

// ---------------------------------------------------------------------------
// Kernel 2: P(8192x8) = V(8192x2048) x MC^T via V_WMMA_F32_16X16X4_F32.
// One wave = 16 rows x 512-wide K slice (KSPLIT=4 -> 2048 waves).
// Double-buffered GLOBAL_LOAD_ASYNC_TO_LDS_B128 streams 16x64 V tiles into
// wave-private LDS (s_wait_asynccnt 8 overlaps next tile with compute); the
// vectors passthrough copy is replayed from LDS with nontemporal stores, so
// V is read from global exactly once. All 16 A/B fragments of a chunk are
// preloaded into registers, then 16 WMMAs issue back-to-back.
//
// Fragment layouts per ISA 7.12.2 (wave32):
//   A 16x4 : lane L -> M = L&15 ; v0 = A[M][2*(L>>4)], v1 = A[M][2*(L>>4)+1]
//   B 4x16 : lane L -> N = L&15 ; v0 = B[2*(L>>4)][N], v1 = B[2*(L>>4)+1][N]
//   D 16x16: lane L, vgpr r -> D[r + 8*(L>>4)][L&15]
// MC is padded to 16 rows (8..15 zero) so B loads are branch-free.
// ---------------------------------------------------------------------------
#define KSPLIT 4
#define KRANGE (FDIM / KSPLIT)   // 512
#define KCHUNK 64
#define NSTEP  (KCHUNK / 4)      // 16 WMMA steps per chunk
#define LSTR   68                // 64+4 pad: frag b64 reads hit 64 distinct banks
#define WAVES  8

__global__ __launch_bounds__(256) void k_pgemm(const float* __restrict__ vec,
                                               float* __restrict__ outvec,
                                               float* __restrict__ ws) {
    __shared__ __align__(16) float lds[2 * WAVES * 16 * LSTR];
    const int tid  = threadIdx.x;
    const int wv   = tid >> 5;
    const int lane = tid & 31;
    const int half = lane >> 4;
    const int l15  = lane & 15;
    const int g    = blockIdx.x * WAVES + wv;   // 0..2047
    const int rt   = g >> 2;                    // row tile 0..511
    const int ks   = g & 3;                     // K split 0..3
    const int rowbase = rt * 16;
    const int k0 = ks * KRANGE;

    float* const wlb0 = lds + (size_t)wv * (16 * LSTR);
    float* const wlb1 = lds + (size_t)(WAVES + wv) * (16 * LSTR);
    const float* mcrow = ws + MC_OFF + (size_t)l15 * FDIM + k0;  // rows 8..15 = zeros

#if HAVE_ASYNC
    auto issue_tile = [&](int kcn, float* w) {
        #pragma unroll
        for (int rr = 0; rr < 8; ++rr) {
            int rl = rr * 2 + half;
            const float* src = vec + (size_t)(rowbase + rl) * FDIM + (size_t)(k0 + kcn + l15 * 4);
            AS1 vsi4* gp = (AS1 vsi4*)(uintptr_t)(const void*)src;
            AS3 vsi4* lp = (AS3 vsi4*)(unsigned)(uintptr_t)(void*)(w + rl * LSTR + l15 * 4);
            __builtin_amdgcn_global_load_async_to_lds_b128(gp, lp, 0, 0);
        }
    };
    issue_tile(0, wlb0);
#endif

    v8f acc = {};
    int buf = 0;
    for (int kc = 0; kc < KRANGE; kc += KCHUNK, buf ^= 1) {
        float* wl = buf ? wlb1 : wlb0;
#if HAVE_ASYNC
        if (kc + KCHUNK < KRANGE) {
            issue_tile(kc + KCHUNK, buf ? wlb0 : wlb1);      // prefetch next tile
            asm volatile("s_wait_asynccnt 8" ::: "memory");  // current tile landed
        } else {
            asm volatile("s_wait_asynccnt 0" ::: "memory");
        }
        // passthrough copy: replay tile from LDS -> streaming global stores
        #pragma unroll
        for (int rr = 0; rr < 8; ++rr) {
            int rl = rr * 2 + half;
            v4f t = *(const v4f*)(wl + rl * LSTR + l15 * 4);
            __builtin_nontemporal_store(
                t, (v4f*)(outvec + (size_t)(rowbase + rl) * FDIM + (size_t)(k0 + kc + l15 * 4)));
        }
#else
        // fallback: bulk register staging for memory-level parallelism
        v4f vv[8];
        #pragma unroll
        for (int rr = 0; rr < 8; ++rr) {
            int rl = rr * 2 + half;
            vv[rr] = *(const v4f*)(vec + (size_t)(rowbase + rl) * FDIM + (size_t)(k0 + kc + l15 * 4));
        }
        #pragma unroll
        for (int rr = 0; rr < 8; ++rr) {
            int rl = rr * 2 + half;
            __builtin_nontemporal_store(
                vv[rr], (v4f*)(outvec + (size_t)(rowbase + rl) * FDIM + (size_t)(k0 + kc + l15 * 4)));
        }
        #pragma unroll
        for (int rr = 0; rr < 8; ++rr) {
            int rl = rr * 2 + half;
            *(v4f*)(wl + rl * LSTR + l15 * 4) = vv[rr];
        }
#endif
        // preload all fragments of this chunk, then WMMA chain back-to-back
        v2f af[NSTEP], bf[NSTEP];
        #pragma unroll
        for (int s = 0; s < NSTEP; ++s) {
            af[s] = *(const v2f*)(wl + l15 * LSTR + 4 * s + 2 * half);
            bf[s] = *(const v2f*)(mcrow + (size_t)(kc + 4 * s + 2 * half));
        }
        #pragma unroll
        for (int s = 0; s < NSTEP; ++s) {
            acc = __builtin_amdgcn_wmma_f32_16x16x4_f32(
                false, af[s], false, bf[s], (short)0, acc, false, false);
        }
    }
    // accumulate K-split partials into P
    if (l15 < 8) {
        float* p = ws + P_OFF;
        #pragma unroll
        for (int r = 0; r < 8; ++r) {
            int grow = rowbase + r + 8 * half;
            atomicAdd(&p[(size_t)grow * 8 + l15], acc[r]);
        }
    }
}

// ---------------------------------------------------------------------------
// Kernel 3: connections[b,c,j,i] = sigmoid(P[b,i,c] + P[b,j,4+c] + const[c]).
// Block = (b, c, 64-wide j tile): p1 row + shifted p2 values in LDS, then
// 64 x 1024 sigmoids as coalesced nontemporal float4 stores (128 MiB stream).
// ---------------------------------------------------------------------------
__device__ __forceinline__ float sigf(float x) {
    float e = __builtin_amdgcn_exp2f(-x * 1.4426950408889634f);
    return __builtin_amdgcn_rcpf(1.0f + e);
}

__global__ __launch_bounds__(256) void k_conn(const float* __restrict__ ws,
                                              float* __restrict__ out) {
    __shared__ float s1[SEQ];
    __shared__ float s2[64];
    const int id  = blockIdx.x;        // 512 blocks
    const int jt  = id & 15;
    const int c   = (id >> 4) & 3;
    const int b   = id >> 6;
    const int tid = threadIdx.x;
    const float* p = ws + P_OFF;
    const float cst = ws[CONST_OFF + c];

    for (int i = tid; i < SEQ; i += 256)
        s1[i] = p[(size_t)((b << 10) + i) * 8 + c];
    if (tid < 64) {
        int j = (jt << 6) + tid;
        s2[tid] = p[(size_t)((b << 10) + j) * 8 + 4 + c] + cst;
    }
    __syncthreads();

    float* obase = out + VEC_ELEMS +
                   ((size_t)((b * CDIM + c) << 10) + (size_t)(jt << 6)) * SEQ;
    const int i0 = tid * 4;
    const float x0 = s1[i0 + 0], x1 = s1[i0 + 1], x2 = s1[i0 + 2], x3 = s1[i0 + 3];
#pragma unroll 4
    for (int jj = 0; jj < 64; ++jj) {
        float s = s2[jj];
        v4f o;
        o.x = sigf(x0 + s);
        o.y = sigf(x1 + s);
        o.z = sigf(x2 + s);
        o.w = sigf(x3 + s);
        __builtin_nontemporal_store(o, (v4f*)(obase + (size_t)jj * SEQ + i0));
    }
}

// ---------------------------------------------------------------------------
extern "C" void kernel_launch(void* const* d_in, const int* in_sizes, int n_in,
                              void* d_out, int out_size, void* d_ws, size_t ws_size,
                              hipStream_t stream) {
    const float* vec = (const float*)d_in[0];   // (8,1024,2048)
    const float* W1  = (const float*)d_in[1];   // (1024,4096)
    const float* b1  = (const float*)d_in[2];   // (1024,)
    const float* W2  = (const float*)d_in[3];   // (4,1024)
    const float* b2  = (const float*)d_in[4];   // (4,)
    float* out = (float*)d_out;                 // [vectors | connections]
    float* ws  = (float*)d_ws;                  // needs ~394 KB

    (void)in_sizes; (void)n_in; (void)out_size; (void)ws_size;

    k_init <<<256, 256, 0, stream>>>(ws, W2, b1, b2);
    k_mc   <<<64,  256, 0, stream>>>(ws, W1, W2);
    k_pgemm<<<256, 256, 0, stream>>>(vec, out, ws);
    k_conn <<<512, 256, 0, stream>>>(ws, out);
}